// RelationPredictor_56839597195462
// MI455X (gfx1250) — compile-verified
//
#include <hip/hip_runtime.h>
#include <cmath>

#define Bsz 64
#define Lsz 2048
#define Dsz 1024
#define Psz 512
#define Csz 32

typedef __attribute__((ext_vector_type(2))) float v2f;
typedef __attribute__((ext_vector_type(8))) float v8f;

__device__ __forceinline__ v8f wmma_f32_k4(v2f a, v2f b, v8f c) {
  // D = A(16x4) * B(4x16) + C, all f32 — exact-precision path for f32 GEMM
  return __builtin_amdgcn_wmma_f32_16x16x4_f32(
      /*neg_a=*/false, a, /*neg_b=*/false, b,
      /*c_mod=*/(short)0, c, /*reuse_a=*/false, /*reuse_b=*/false);
}

__global__ __launch_bounds__(512, 1) void relpred_fused(
    const float* __restrict__ enc, const int* __restrict__ prep_idx,
    const float* __restrict__ Wh, const float* __restrict__ Wp,
    const float* __restrict__ Wc, const float* __restrict__ Whid,
    const float* __restrict__ Wscore, float* __restrict__ out)
{
  __shared__ float comp1[16 * Psz];   // tanh(3-way GEMM) tile, 32 KB
  __shared__ float comp2[16 * Psz];   // tanh(hidden GEMM) tile, 32 KB

  const int tid  = threadIdx.x;
  const int lane = tid & 31;
  const int wave = tid >> 5;          // 0..15
  const int half = (lane >> 4) & 1;   // K-half for A/B operands
  const int mr   = lane & 15;         // M row (A/C) / N col (B/C)
  const int m0   = blockIdx.x * 16;   // batch tile
  const int n0   = wave * 32;         // this wave's N base (two 16-wide tiles)

  // ---------------- stage 1: tanh(head@Wh + prep@Wp + child@Wc) ----------
  const int bidx = m0 + mr;
  const int prep = prep_idx[bidx];

  v8f acc_a = {}; v8f acc_b = {};
  #pragma unroll
  for (int seg = 0; seg < 3; ++seg) {
    const float* wseg = (seg == 0) ? Wh : (seg == 1) ? Wp : Wc;
    const float* arow = enc + ((size_t)bidx * Lsz + (size_t)(prep - 1 + seg)) * Dsz
                            + half * 2;
    #pragma unroll 4
    for (int k = 0; k < Dsz; k += 4) {
      v2f a;  a[0] = arow[k];  a[1] = arow[k + 1];
      const float* wr0 = wseg + (size_t)(k + half * 2) * Psz + n0 + mr;
      const float* wr1 = wr0 + Psz;
      v2f ba; ba[0] = wr0[0];  ba[1] = wr1[0];
      v2f bb; bb[0] = wr0[16]; bb[1] = wr1[16];
      acc_a = wmma_f32_k4(a, ba, acc_a);
      acc_b = wmma_f32_k4(a, bb, acc_b);
    }
  }
  #pragma unroll
  for (int i = 0; i < 8; ++i) {
    const int m = i + half * 8;
    comp1[m * Psz + n0 + mr]      = tanhf(acc_a[i]);
    comp1[m * Psz + n0 + 16 + mr] = tanhf(acc_b[i]);
  }
  __syncthreads();

  // ---------------- stage 2: tanh(composed @ hidden_w) -------------------
  v8f acc2a = {}; v8f acc2b = {};
  {
    const float* arow = comp1 + mr * Psz + half * 2;
    #pragma unroll 4
    for (int k = 0; k < Psz; k += 4) {
      v2f a;  a[0] = arow[k];  a[1] = arow[k + 1];
      const float* wr0 = Whid + (size_t)(k + half * 2) * Psz + n0 + mr;
      const float* wr1 = wr0 + Psz;
      v2f ba; ba[0] = wr0[0];  ba[1] = wr1[0];
      v2f bb; bb[0] = wr0[16]; bb[1] = wr1[16];
      acc2a = wmma_f32_k4(a, ba, acc2a);
      acc2b = wmma_f32_k4(a, bb, acc2b);
    }
  }
  #pragma unroll
  for (int i = 0; i < 8; ++i) {
    const int m = i + half * 8;
    comp2[m * Psz + n0 + mr]      = tanhf(acc2a[i]);
    comp2[m * Psz + n0 + 16 + mr] = tanhf(acc2b[i]);
  }
  __syncthreads();

  // ---------------- stage 3: softmax(composed2 @ scorer), wave 0 ---------
  if (wave == 0) {   // wave-uniform branch; EXEC all-ones inside
    v8f s0 = {}; v8f s1 = {};
    const float* arow = comp2 + mr * Psz + half * 2;
    #pragma unroll 4
    for (int k = 0; k < Psz; k += 4) {
      v2f a;  a[0] = arow[k];  a[1] = arow[k + 1];
      const float* wr0 = Wscore + (size_t)(k + half * 2) * Csz + mr;
      const float* wr1 = wr0 + Csz;
      v2f ba; ba[0] = wr0[0];  ba[1] = wr1[0];
      v2f bb; bb[0] = wr0[16]; bb[1] = wr1[16];
      s0 = wmma_f32_k4(a, ba, s0);
      s1 = wmma_f32_k4(a, bb, s1);
    }
    // each output row's 32 scores live across the 16 lanes of one half
    // (two accumulator VGPRs per lane); xor masks 1..8 stay inside a half.
    #pragma unroll
    for (int i = 0; i < 8; ++i) {
      const int m = i + half * 8;
      float v0 = s0[i], v1 = s1[i];
      float mx = fmaxf(v0, v1);
      for (int off = 8; off >= 1; off >>= 1)
        mx = fmaxf(mx, __shfl_xor(mx, off, 32));
      float e0 = expf(v0 - mx), e1 = expf(v1 - mx);
      float sum = e0 + e1;
      for (int off = 8; off >= 1; off >>= 1)
        sum += __shfl_xor(sum, off, 32);
      const float inv = 1.0f / sum;
      out[(size_t)(m0 + m) * Csz + mr]      = e0 * inv;
      out[(size_t)(m0 + m) * Csz + 16 + mr] = e1 * inv;
    }
  }
}

extern "C" void kernel_launch(void* const* d_in, const int* in_sizes, int n_in,
                              void* d_out, int out_size, void* d_ws, size_t ws_size,
                              hipStream_t stream) {
  (void)in_sizes; (void)n_in; (void)d_ws; (void)ws_size; (void)out_size;
  const float* enc    = (const float*)d_in[0];
  const int*   pidx   = (const int*)  d_in[1];
  const float* Wh     = (const float*)d_in[2];
  const float* Wp     = (const float*)d_in[3];
  const float* Wc     = (const float*)d_in[4];
  const float* Whid   = (const float*)d_in[5];
  const float* Wscore = (const float*)d_in[6];
  relpred_fused<<<Bsz / 16, 512, 0, stream>>>(enc, pidx, Wh, Wp, Wc, Whid,
                                              Wscore, (float*)d_out);
}